// ExpansionFX_25451976196770
// MI455X (gfx1250) — compile-verified
//
#include <hip/hip_runtime.h>
#include <hip/hip_bf16.h>
#include <cstddef>

// ---------------------------------------------------------------------------
// ExpansionFX on MI455X (gfx1250, wave32).
//
// HBM-bound: weights stream (633 MB of 766 MB total) dominates => ~33 us
// floor at 23.3 TB/s. Everything fused in LDS; weights read exactly once via
// b128 loads; the shared Wigner-3j contraction runs on V_WMMA_F32_16X16X4_F32
// with the C/D operand serving as the cross-path accumulator. All WMMA
// fragment traffic is branch-free (zero-padded LDS staging) so EXEC stays
// all-ones and no exec-mask save/restore appears around ds loads.
// ---------------------------------------------------------------------------

typedef float v2f __attribute__((ext_vector_type(2)));
typedef float v8f __attribute__((ext_vector_type(8)));

#define NPATH 15

// Per-path constants (order = reference _instructions() order).
__device__ __constant__ int   I_mul [NPATH] = {32,32,32, 16,16,16,16,16,16, 8,8,8,8,8,8};
__device__ __constant__ int   I_n1  [NPATH] = {16, 8, 4, 16, 8, 8, 8, 4, 4, 16, 8, 8, 4, 4, 4};
__device__ __constant__ int   I_n2  [NPATH] = {16, 8, 4,  8,16, 8, 4, 8, 4,  4, 8, 4,16, 8, 4};
__device__ __constant__ int   I_din [NPATH] = { 1, 1, 1,  3, 3, 3, 3, 3, 3,  5, 5, 5, 5, 5, 5};
__device__ __constant__ int   I_d1  [NPATH] = { 1, 3, 5,  1, 3, 3, 3, 5, 5,  1, 3, 3, 5, 5, 5};
__device__ __constant__ int   I_d2  [NPATH] = { 1, 3, 5,  3, 1, 3, 5, 3, 5,  5, 3, 5, 1, 3, 5};
__device__ __constant__ int   I_woff[NPATH] = {0,8192,10240,10752,12800,14848,15872,16384,16896,
                                               17152,17664,18176,18432,18944,19200};
__device__ __constant__ int   I_joff[NPATH] = {0,1,10,35,44,53,80,125,170,245,270,315,390,415,490};
__device__ __constant__ int   I_boff[NPATH] = {0,256,320, -1,-1,-1,-1,-1,-1, -1,-1,-1,-1,-1,-1};
__device__ __constant__ int   I_xoff[NPATH] = {0,0,0, 32,32,32,32,32,32, 80,80,80,80,80,80};
// Accumulator-block base (output block (j,k), layout [uv][d1*d2]).
__device__ __constant__ int   I_acc [NPATH] = {0,1344,3200, 256,960,1344,1920,2720,3200,
                                               640,1344,1920,2400,2720,3200};
__device__ __constant__ float I_rmul[NPATH] = {0.03125f,0.03125f,0.03125f,
                                               0.0625f,0.0625f,0.0625f,0.0625f,0.0625f,0.0625f,
                                               0.125f,0.125f,0.125f,0.125f,0.125f,0.125f};
// Writeout decode tables.
__device__ __constant__ int ACC_BASE[9] = {0,256,640, 960,1344,1920, 2400,2720,3200}; // [j*3+k]
__device__ __constant__ int O2_N[3] = {16,8,4};
__device__ __constant__ int O2_D[3] = {1,3,5};
__device__ __constant__ int O1_D[3] = {1,3,5};

__global__ __launch_bounds__(128)
void expansion_fx_kernel(const float* __restrict__ x_in,
                         const float* __restrict__ weights,
                         const float* __restrict__ bias_w,
                         const float* __restrict__ w3j_flat,
                         float* __restrict__ out)
{
    const int b    = blockIdx.x;
    const int tid  = threadIdx.x;
    const int lane = tid & 31;
    const int wave = tid >> 5;

    __shared__ float xsL[120];        // staged input irreps
    __shared__ float w3jL[615];       // staged Wigner-3j
    __shared__ float accL[3600];      // output accumulator, per-block [uv][ij]
    __shared__ float resL[8 * 256];   // per-path res, k-major, zero-padded K rows
    __shared__ float BmatL[8 * 32];   // per-path dense, zero-padded B (w3j) matrix
    __shared__ float dumpL[16];       // sink for out-of-range D-store lanes

    const float* xb = x_in    + (size_t)b * 120;
    const float* Wb = weights + (size_t)b * 19328;
    const float* Bb = bias_w  + (size_t)b * 336;

    for (int t = tid; t < 120;      t += 128) xsL[t]  = xb[t];
    for (int t = tid; t < 615;      t += 128) w3jL[t] = w3j_flat[t];
    for (int t = tid; t < 3600;     t += 128) accL[t] = 0.0f;
    for (int t = tid; t < 8 * 256;  t += 128) resL[t] = 0.0f;   // K-padding invariant
    __syncthreads();

    for (int p = 0; p < NPATH; ++p) {
        const int n1   = I_n1[p];
        const int n2   = I_n2[p];
        const int din  = I_din[p];
        const int d1   = I_d1[p];
        const int d2   = I_d2[p];
        const int mul  = I_mul[p];
        const int n1n2 = n1 * n2;
        const int ij   = d1 * d2;
        const float rmul = I_rmul[p];
        const float* __restrict__ Wp = Wb + I_woff[p];
        const int xoff = I_xoff[p];
        const int boff = I_boff[p];
        const int jo   = I_joff[p];
        const int ktiles = (din + 3) >> 2;

        // Warm the next path's weight stream (global_prefetch_b8, speculative).
        if (p + 1 < NPATH)
            __builtin_prefetch(Wb + I_woff[p + 1] + tid * 16, 0, 0);

        // ---- Build zero-padded dense B matrix: BmatL[k][n] = w3j[n][k] ----
        for (int idx = tid; idx < (ktiles << 7); idx += 128) {
            const int kk = idx >> 5;
            const int nn = idx & 31;
            BmatL[idx] = (kk < din && nn < ij) ? w3jL[jo + nn * din + kk] : 0.0f;
        }

        // ---- Phase 1: res[k][uv] = (1/mul) * (bias + sum_w W[w,uv] * x[w,k]) ----
        // b128 weight loads, coalesced across lanes, each weight read once.
        // All 5 K rows stored unconditionally (zero beyond din) so the WMMA
        // A-fragment reads below need no predication.
        {
            const int ncell4 = n1n2 >> 2;                    // n1n2 % 4 == 0 always
            const float4* __restrict__ W4 = reinterpret_cast<const float4*>(Wp);
            for (int c4 = tid; c4 < ncell4; c4 += 128) {
                float rk[4][5];
                #pragma unroll
                for (int q = 0; q < 4; ++q)
                    #pragma unroll
                    for (int k = 0; k < 5; ++k) rk[q][k] = 0.0f;

                for (int w = 0; w < mul; ++w) {
                    const float4 wv = W4[w * ncell4 + c4];
                    #pragma unroll
                    for (int k = 0; k < 5; ++k) {
                        // For k >= din this reads a valid (finite) but unused
                        // xsL entry; the result is zero-selected at the store.
                        const float xv = xsL[xoff + w * din + k];
                        rk[0][k] = fmaf(wv.x, xv, rk[0][k]);
                        rk[1][k] = fmaf(wv.y, xv, rk[1][k]);
                        rk[2][k] = fmaf(wv.z, xv, rk[2][k]);
                        rk[3][k] = fmaf(wv.w, xv, rk[3][k]);
                    }
                }
                if (boff >= 0) {                              // scalar paths: din == 1
                    const float4 bb = reinterpret_cast<const float4*>(Bb + boff)[c4];
                    rk[0][0] += bb.x; rk[1][0] += bb.y;
                    rk[2][0] += bb.z; rk[3][0] += bb.w;
                }
                #pragma unroll
                for (int k = 0; k < 5; ++k) {
                    float4 st;
                    st.x = (k < din) ? rk[0][k] * rmul : 0.0f;
                    st.y = (k < din) ? rk[1][k] * rmul : 0.0f;
                    st.z = (k < din) ? rk[2][k] * rmul : 0.0f;
                    st.w = (k < din) ? rk[3][k] * rmul : 0.0f;
                    reinterpret_cast<float4*>(resL + k * 256)[c4] = st;
                }
            }
        }
        __syncthreads();

        // ---- Phase 2: accL[uv][n] += sum_k res[uv][k] * w3j[n][k] via WMMA ----
        // GEMM: M = n1*n2 (multiple of 16), K = din padded to 4*ktiles,
        // N = d1*d2 padded to 16*ntiles. Fragment traffic fully branch-free.
        const int mtiles = n1n2 >> 4;
        const int ntiles = (ij + 15) >> 4;
        const int tiles  = mtiles * ntiles;
        float* __restrict__ acc = accL + I_acc[p];

        for (int t = wave; t < tiles; t += 4) {               // wave-uniform loop
            const int m0 = (t / ntiles) << 4;
            const int n0 = (t % ntiles) << 4;
            const int ncol  = n0 + (lane & 15);
            const bool nval = ncol < ij;
            const int ccol  = nval ? ncol : 0;                // clamped, always valid

            // C/D 16x16 f32 layout: VGPR r, lanes 0-15 -> M=m0+r, lanes 16-31 -> M=m0+8+r.
            v8f c;
            #pragma unroll
            for (int r = 0; r < 8; ++r) {
                const int mrow = m0 + r + ((lane >> 4) << 3);
                const float cv = acc[mrow * ij + ccol];       // unconditional ds_load
                c[r] = nval ? cv : 0.0f;
            }

            for (int kt = 0; kt < ktiles; ++kt) {
                const int kb = kt << 2;
                // A 16x4 f32 layout: lanes 0-15 hold K=kb,kb+1; lanes 16-31 K=kb+2,kb+3.
                const int mrowA = m0 + (lane & 15);
                const int ka    = kb + ((lane >> 4) << 1);
                v2f a;
                a[0] = resL[ ka      * 256 + mrowA];          // zero-padded, no guard
                a[1] = resL[(ka + 1) * 256 + mrowA];
                // B 4x16 f32 layout: VGPR0 rows K=kb(lo)/kb+1(hi), VGPR1 K=kb+2/kb+3.
                const int kb0 = kb + (lane >> 4);
                v2f bv;
                bv[0] = BmatL[ kb0      * 32 + ncol];         // zero-padded, no guard
                bv[1] = BmatL[(kb0 + 2) * 32 + ncol];

                c = __builtin_amdgcn_wmma_f32_16x16x4_f32(
                        /*neg_a=*/false, a, /*neg_b=*/false, bv,
                        /*c_mod=*/(short)0, c, /*reuse_a=*/false, /*reuse_b=*/false);
            }

            #pragma unroll
            for (int r = 0; r < 8; ++r) {
                const int mrow = m0 + r + ((lane >> 4) << 3);
                float* dst = nval ? (acc + mrow * ij + ncol) : (dumpL + (lane & 15));
                *dst = c[r];                                   // unconditional ds_store
            }
        }
        __syncthreads();
    }

    // ---- Writeout: decode 60x60 output coordinates, coalesced stores ----
    float* ob = out + (size_t)b * 3600;
    for (int idx = tid; idx < 3600; idx += 128) {
        const int row = idx / 60;
        const int col = idx % 60;
        int j, u, ii;
        if (row < 16)      { j = 0; u = row;            ii = 0; }
        else if (row < 40) { j = 1; u = (row - 16) / 3; ii = (row - 16) % 3; }
        else               { j = 2; u = (row - 40) / 5; ii = (row - 40) % 5; }
        int k, v, jj;
        if (col < 16)      { k = 0; v = col;            jj = 0; }
        else if (col < 40) { k = 1; v = (col - 16) / 3; jj = (col - 16) % 3; }
        else               { k = 2; v = (col - 40) / 5; jj = (col - 40) % 5; }
        const int ld = O1_D[j] * O2_D[k];
        ob[idx] = accL[ACC_BASE[j * 3 + k] + (u * O2_N[k] + v) * ld + (ii * O2_D[k] + jj)];
    }
}

extern "C" void kernel_launch(void* const* d_in, const int* in_sizes, int n_in,
                              void* d_out, int out_size, void* d_ws, size_t ws_size,
                              hipStream_t stream) {
    (void)n_in; (void)d_ws; (void)ws_size; (void)in_sizes;
    const float* x_in     = (const float*)d_in[0];
    const float* weights  = (const float*)d_in[1];
    const float* bias_w   = (const float*)d_in[2];
    const float* w3j_flat = (const float*)d_in[3];
    float* out = (float*)d_out;

    const int B = out_size / 3600;   // 8192
    hipLaunchKernelGGL(expansion_fx_kernel, dim3(B), dim3(128), 0, stream,
                       x_in, weights, bias_w, w3j_flat, out);
}